// SAGMM_network_1623497638192
// MI455X (gfx1250) — compile-verified
//
#include <hip/hip_runtime.h>
#include <hip/hip_bf16.h>
#include <math.h>

// ---------------- problem constants (match reference) ----------------
#define NNODES 20000
#define NEDGES 160000
#define DIM    128     // D == H == C
#define NEXP   4
#define COEF   0.01f
#define ROWTILES (NNODES / 16)   // 1250

typedef __attribute__((ext_vector_type(2))) float v2f;
typedef __attribute__((ext_vector_type(8))) float v8f;
typedef __attribute__((ext_vector_type(4))) unsigned int u32x4;
typedef __attribute__((ext_vector_type(4))) int i32x4;
typedef __attribute__((ext_vector_type(8))) int i32x8;

// ---------------- TDM: stage the 128x128 f32 weight matrix into LDS --------
// D# descriptor per CDNA5 ISA ch.8: 2D tensor, data_size=4B,
// tensor_dim0=tensor_dim1=128, tile_dim0=tile_dim1=128, dim0_stride=128.
// Issued by one wave per block (TDM ignores EXEC; per-wave op, TENSORcnt).
__device__ __forceinline__ void tdm_load_B_to_lds(const float* gB, unsigned lds_off) {
  unsigned long long ga = (unsigned long long)(size_t)gB;
  u32x4 g0;
  g0.x = 1u;                                            // count=1, user descriptor
  g0.y = lds_off;                                       // lds_addr [63:32]
  g0.z = (unsigned)(ga & 0xffffffffu);                  // global_addr [95:64]
  g0.w = (unsigned)((ga >> 32) & 0x1ffffffu) | (2u << 30); // addr hi + type=2
  i32x8 g1;
  g1[0] = (int)(2u << 16);      // workgroup_mask=0, data_size=2 (4 bytes)
  g1[1] = (int)(128u << 16);    // atomic_barrier_addr=0 | tensor_dim0.lo=128
  g1[2] = (int)(128u << 16);    // tensor_dim0.hi=0 | tensor_dim1.lo=128
  g1[3] = (int)(128u << 16);    // tensor_dim1.hi=0 | tile_dim0=128
  g1[4] = 128;                  // tile_dim1=128, tile_dim2=0
  g1[5] = 128;                  // tensor_dim0_stride.lo=128
  g1[6] = 0;                    // stride0.hi=0 | dim1_stride.lo=0
  g1[7] = 0;
  i32x4 z4 = {0, 0, 0, 0};
#if __clang_major__ >= 23
  i32x8 z8 = {0, 0, 0, 0, 0, 0, 0, 0};
  __builtin_amdgcn_tensor_load_to_lds(g0, g1, z4, z4, z8, 0);
#else
  __builtin_amdgcn_tensor_load_to_lds(g0, g1, z4, z4, 0);
#endif
}

// ---------------- utility ----------------
__global__ void zero_kernel(float* __restrict__ p, int n) {
  int i = blockIdx.x * blockDim.x + threadIdx.x;
  if (i < n) p[i] = 0.f;
}

__global__ void deg_kernel(const int* __restrict__ dst, float* __restrict__ deg, int E) {
  int i = blockIdx.x * blockDim.x + threadIdx.x;
  if (i < E) atomicAdd(&deg[dst[i]], 1.f);
}

__global__ void norm_kernel(const int* __restrict__ src, const int* __restrict__ dst,
                            const float* __restrict__ deg, float* __restrict__ norm, int E) {
  int i = blockIdx.x * blockDim.x + threadIdx.x;
  if (i < E) {
    float a = fmaxf(deg[src[i]], 1.f);
    float b = fmaxf(deg[dst[i]], 1.f);
    norm[i] = rsqrtf(a * b);
  }
}

// gates[n,e] = (sigmoid(x_agg[n,:]@w_gate[:,e]) - thr[e] > 0 ? 1 : 0) * mask[e]
__global__ void gates_kernel(const float* __restrict__ xagg, const float* __restrict__ wg,
                             const float* __restrict__ thr, const float* __restrict__ mask,
                             float* __restrict__ gates, float* __restrict__ importance, int n) {
  int i = blockIdx.x * blockDim.x + threadIdx.x;
  bool active = (i < n);
  float acc0 = 0.f, acc1 = 0.f, acc2 = 0.f, acc3 = 0.f;
  if (active) {
    const float* xr = xagg + (size_t)i * DIM;
    #pragma unroll 4
    for (int d = 0; d < DIM; ++d) {
      float xv = xr[d];
      const float* w = wg + d * NEXP;
      acc0 += xv * w[0]; acc1 += xv * w[1]; acc2 += xv * w[2]; acc3 += xv * w[3];
    }
  }
  float accs[NEXP] = {acc0, acc1, acc2, acc3};
  #pragma unroll
  for (int e = 0; e < NEXP; ++e) {
    float gv = 0.f;
    if (active) {
      float s = 1.f / (1.f + __expf(-accs[e]));
      gv = ((s - thr[e]) > 0.f ? 1.f : 0.f) * mask[e];
      gates[(size_t)i * NEXP + e] = gv;
    }
    float red = gv;                     // wave32 reduction before atomics
    for (int off = 16; off > 0; off >>= 1) red += __shfl_xor(red, off, 32);
    if ((threadIdx.x & 31) == 0 && red != 0.f) atomicAdd(&importance[e], red);
  }
}

// Edge scatter: out[dst,:] += feat[src,:] * norm[e].  One wave per edge,
// one float4 per lane (32 lanes * 4 = 128 features, coalesced 512B/wave).
__global__ void scatter_kernel(const float* __restrict__ feat, const int* __restrict__ src,
                               const int* __restrict__ dst, const float* __restrict__ norm,
                               float* __restrict__ out, int E) {
  int t = blockIdx.x * blockDim.x + threadIdx.x;
  int e = t >> 5;
  int lane = t & 31;
  if (e >= E) return;
  int s = src[e];
  int d = dst[e];
  float w = norm[e];
  const float4* fs = (const float4*)(feat + (size_t)s * DIM);
  float4 v = fs[lane];
  float* od = out + (size_t)d * DIM + lane * 4;
  atomicAdd(od + 0, v.x * w);
  atomicAdd(od + 1, v.y * w);
  atomicAdd(od + 2, v.z * w);
  atomicAdd(od + 3, v.w * w);
}

// ---------------- WMMA GEMMs (exact f32: v_wmma_f32_16x16x4_f32) -----------
// Block = 4 waves. TDM stages B (64KB) into LDS once per block; each wave
// computes a 16x128 output strip: 8 accumulators, A-fragment reused across
// the 8 column tiles, B fragments read from LDS.
// A-frag (16x4 f32): lane m=lane%16; VGPR0/1 = K pair {k0,k0+1} (lanes 0-15)
//                    or {k0+2,k0+3} (lanes 16-31)   [ISA 7.12.2]
// C/D   (16x16 f32): 8 VGPRs, M = r + 8*(lane>=16), N = lane%16

// h = relu(A @ W1e + b1e)
__global__ __launch_bounds__(128) void gemm_relu_kernel(
    const float* __restrict__ A, const float* __restrict__ B,
    const float* __restrict__ bias, float* __restrict__ out) {
  __shared__ float ldsB[DIM * DIM];          // 64 KB of 320 KB/WGP
  const int lane = threadIdx.x & 31;
  const int wv   = threadIdx.x >> 5;
  if (wv == 0) {
    tdm_load_B_to_lds(B, (unsigned)(size_t)&ldsB[0]);
    __builtin_amdgcn_s_wait_tensorcnt(0);
  }
  __syncthreads();
  const int rowTile = blockIdx.x * 4 + wv;
  if (rowTile < ROWTILES) {                  // wave-uniform -> EXEC all-1 below
    const int row0 = rowTile << 4;
    const int nn   = lane & 15;
    const int half = lane >> 4;
    const float* arow = A + (size_t)(row0 + nn) * DIM;
    v8f acc[8] = {};
    for (int k0 = 0; k0 < DIM; k0 += 4) {
      const int kk = k0 + (half << 1);
      v2f a;
      a.x = arow[kk];
      a.y = arow[kk + 1];
      #pragma unroll
      for (int c = 0; c < 8; ++c) {
        v2f b;
        b.x = ldsB[kk * DIM + (c << 4) + nn];
        b.y = ldsB[(kk + 1) * DIM + (c << 4) + nn];
        acc[c] = __builtin_amdgcn_wmma_f32_16x16x4_f32(false, a, false, b,
                                                       (short)0, acc[c], false, false);
      }
    }
    #pragma unroll
    for (int c = 0; c < 8; ++c) {
      const int col0 = c << 4;
      const float bv = bias[col0 + nn];
      #pragma unroll
      for (int r = 0; r < 8; ++r) {
        float v = fmaxf(acc[c][r] + bv, 0.f);
        out[(size_t)(row0 + r + (half << 3)) * DIM + col0 + nn] = v;
      }
    }
  }
}

// out[n,c] += gates[n,expert] * (A @ W2e + b2e)[n,c]
__global__ __launch_bounds__(128) void gemm_combine_kernel(
    const float* __restrict__ A, const float* __restrict__ B,
    const float* __restrict__ bias, const float* __restrict__ gates,
    int expert, float* __restrict__ out) {
  __shared__ float ldsB[DIM * DIM];
  const int lane = threadIdx.x & 31;
  const int wv   = threadIdx.x >> 5;
  if (wv == 0) {
    tdm_load_B_to_lds(B, (unsigned)(size_t)&ldsB[0]);
    __builtin_amdgcn_s_wait_tensorcnt(0);
  }
  __syncthreads();
  const int rowTile = blockIdx.x * 4 + wv;
  if (rowTile < ROWTILES) {
    const int row0 = rowTile << 4;
    const int nn   = lane & 15;
    const int half = lane >> 4;
    const float* arow = A + (size_t)(row0 + nn) * DIM;
    v8f acc[8] = {};
    for (int k0 = 0; k0 < DIM; k0 += 4) {
      const int kk = k0 + (half << 1);
      v2f a;
      a.x = arow[kk];
      a.y = arow[kk + 1];
      #pragma unroll
      for (int c = 0; c < 8; ++c) {
        v2f b;
        b.x = ldsB[kk * DIM + (c << 4) + nn];
        b.y = ldsB[(kk + 1) * DIM + (c << 4) + nn];
        acc[c] = __builtin_amdgcn_wmma_f32_16x16x4_f32(false, a, false, b,
                                                       (short)0, acc[c], false, false);
      }
    }
    #pragma unroll
    for (int r = 0; r < 8; ++r) {
      const int row = row0 + r + (half << 3);
      const float g = gates[(size_t)row * NEXP + expert];
      #pragma unroll
      for (int c = 0; c < 8; ++c) {
        const int col0 = c << 4;
        float* op = out + (size_t)row * DIM + col0 + nn;
        *op += g * (acc[c][r] + bias[col0 + nn]);  // experts stream-sequenced
      }
    }
  }
}

// load_loss = COEF * var(importance) / (mean(importance)^2 + 1e-10)
__global__ void loss_kernel(const float* __restrict__ importance, float* __restrict__ out_loss) {
  if (threadIdx.x == 0 && blockIdx.x == 0) {
    float m = 0.f;
    for (int e = 0; e < NEXP; ++e) m += importance[e];
    m *= (1.f / NEXP);
    float var = 0.f;
    for (int e = 0; e < NEXP; ++e) {
      float d = importance[e] - m;
      var += d * d;
    }
    var *= (1.f / NEXP);
    out_loss[0] = COEF * (var / (m * m + 1e-10f));
  }
}

// ---------------- driver ----------------
extern "C" void kernel_launch(void* const* d_in, const int* in_sizes, int n_in,
                              void* d_out, int out_size, void* d_ws, size_t ws_size,
                              hipStream_t stream) {
  const float* x      = (const float*)d_in[0];   // [N,D]
  const int*   edge   = (const int*)  d_in[1];   // [2,E]
  const float* x_agg  = (const float*)d_in[2];   // [N,D]
  const float* w_gate = (const float*)d_in[3];   // [D,NE]
  const float* thr    = (const float*)d_in[4];   // [NE]
  const float* mask   = (const float*)d_in[5];   // [NE]
  const float* W1     = (const float*)d_in[6];   // [NE,D,H]
  const float* b1     = (const float*)d_in[7];   // [NE,H]
  const float* W2     = (const float*)d_in[8];   // [NE,H,C]
  const float* b2     = (const float*)d_in[9];   // [NE,C]
  float* out = (float*)d_out;                    // [N*C] output ++ [1] loss

  const int* src = edge;
  const int* dst = edge + NEDGES;

  // workspace layout (floats), ~31.8 MB total, all chunks 16B aligned
  float* ws   = (float*)d_ws;
  float* deg  = ws;                 // 20000
  float* nrm  = ws + 20000;         // 160000
  float* gts  = ws + 180000;        // 80000
  float* imp  = ws + 260000;        // 16 (4 used)
  float* aggx = ws + 260016;        // 2,560,000
  float* h    = ws + 2820016;       // 2,560,000 (reused per expert)
  float* aggh = ws + 5380016;       // 2,560,000 (reused per expert)

  const int NF = NNODES * DIM;      // 2,560,000

  zero_kernel<<<(NNODES + 255) / 256, 256, 0, stream>>>(deg, NNODES);
  zero_kernel<<<1, 32, 0, stream>>>(imp, 16);
  zero_kernel<<<(NF + 255) / 256, 256, 0, stream>>>(out, NF);
  zero_kernel<<<(NF + 255) / 256, 256, 0, stream>>>(aggx, NF);

  deg_kernel <<<(NEDGES + 255) / 256, 256, 0, stream>>>(dst, deg, NEDGES);
  norm_kernel<<<(NEDGES + 255) / 256, 256, 0, stream>>>(src, dst, deg, nrm, NEDGES);
  gates_kernel<<<(NNODES + 127) / 128, 128, 0, stream>>>(x_agg, w_gate, thr, mask, gts, imp, NNODES);
  scatter_kernel<<<(NEDGES * 32 + 255) / 256, 256, 0, stream>>>(x, src, dst, nrm, aggx, NEDGES);

  const int gblocks = (ROWTILES + 3) / 4;   // 313 blocks of 4 waves (128 thr)

  for (int e = 0; e < NEXP; ++e) {
    gemm_relu_kernel<<<gblocks, 128, 0, stream>>>(
        aggx, W1 + (size_t)e * DIM * DIM, b1 + e * DIM, h);
    zero_kernel<<<(NF + 255) / 256, 256, 0, stream>>>(aggh, NF);
    scatter_kernel<<<(NEDGES * 32 + 255) / 256, 256, 0, stream>>>(h, src, dst, nrm, aggh, NEDGES);
    gemm_combine_kernel<<<gblocks, 128, 0, stream>>>(
        aggh, W2 + (size_t)e * DIM * DIM, b2 + e * DIM, gts, e, out);
  }

  loss_kernel<<<1, 1, 0, stream>>>(imp, out + NF);
}